// CrossEntropyWithLabelSmooth_4054449127455
// MI455X (gfx1250) — compile-verified
//
#include <hip/hip_runtime.h>
#include <math.h>
#include <stdint.h>

#define LOG2E 1.4426950408889634f
#define EPS   0.1f

// Native clang vector type so __builtin_nontemporal_load accepts it.
typedef float v4f __attribute__((ext_vector_type(4)));

// Combine two online-softmax partial states (m = running max, s = sum exp(x-m)).
// Guarded so (-inf,-inf) pairs don't produce NaN via (-inf) - (-inf).
__device__ __forceinline__ void combine_ms(float& m, float& s, float mo, float so) {
    float M = fmaxf(m, mo);
    float a = (m  == M) ? s  : s  * exp2f((m  - M) * LOG2E);
    float b = (mo == M) ? so : so * exp2f((mo - M) * LOG2E);
    m = M;
    s = a + b;
}

__device__ __forceinline__ float max4(v4f v) {
    return fmaxf(fmaxf(v.x, v.y), fmaxf(v.z, v.w));
}

__device__ __forceinline__ float exps4(v4f v, float M) {
    return exp2f((v.x - M) * LOG2E) + exp2f((v.y - M) * LOG2E)
         + exp2f((v.z - M) * LOG2E) + exp2f((v.w - M) * LOG2E);
}

// One workgroup (8 wave32s) per row. Single streaming pass over 201 KB/row:
// branchless online max / sum-exp / plain sum with b128 non-temporal loads.
__global__ __launch_bounds__(256) void ce_ls_rows_kernel(
    const float* __restrict__ inputs,
    const int*   __restrict__ targets,
    float*       __restrict__ per_row,
    int K)
{
    const int row  = blockIdx.x;
    const int tid  = threadIdx.x;
    const int nthr = blockDim.x;
    const float* rp = inputs + (size_t)row * (size_t)K;

    float m  = -INFINITY;  // running max
    float s  = 0.0f;       // running sum of exp(x - m)
    float sx = 0.0f;       // running sum of x

    // Peel 0..3 head elements so the vector body is 16B-aligned
    // (K % 4 == 1, so row bases are only 4B-aligned).
    int lead = (int)(((uintptr_t)rp >> 2) & 3u);
    lead = (4 - lead) & 3;
    if (lead > K) lead = K;
    for (int i = tid; i < lead; i += nthr) {
        float x = rp[i];
        sx += x;
        float M = fmaxf(m, x);
        s = s * exp2f((m - M) * LOG2E) + exp2f((x - M) * LOG2E);
        m = M;
    }

    const int  nvec = (K - lead) >> 2;
    const v4f* vp   = (const v4f*)(rp + lead);

    // Hot loop: 4 independent b128 NT loads per iteration (64 B/thread/iter,
    // wave covers 512 B contiguous per load), then one branchless rescale over
    // the 16-element chunk: 17 v_exp_f32 per 16 elements, no inner-loop
    // divergence.
    int i = tid;
    const int step4 = nthr << 2;
    for (; i + 3 * nthr < nvec; i += step4) {
        v4f a = __builtin_nontemporal_load(vp + i);
        v4f b = __builtin_nontemporal_load(vp + i + nthr);
        v4f c = __builtin_nontemporal_load(vp + i + 2 * nthr);
        v4f d = __builtin_nontemporal_load(vp + i + 3 * nthr);

        v4f sv = (a + b) + (c + d);
        sx += (sv.x + sv.y) + (sv.z + sv.w);

        float mv = fmaxf(fmaxf(max4(a), max4(b)), fmaxf(max4(c), max4(d)));
        float M  = fmaxf(m, mv);
        float acc = s * exp2f((m - M) * LOG2E);   // == s when M == m
        acc += exps4(a, M);
        acc += exps4(b, M);
        acc += exps4(c, M);
        acc += exps4(d, M);
        s = acc;
        m = M;
    }
    // Remainder vectors (branchless single-chunk update).
    for (; i < nvec; i += nthr) {
        v4f v = __builtin_nontemporal_load(vp + i);
        sx += (v.x + v.y) + (v.z + v.w);
        float M = fmaxf(m, max4(v));
        s = s * exp2f((m - M) * LOG2E) + exps4(v, M);
        m = M;
    }

    // Tail scalars (K % 4 leftovers).
    for (int j = lead + (nvec << 2) + tid; j < K; j += nthr) {
        float x = rp[j];
        sx += x;
        float M = fmaxf(m, x);
        s = s * exp2f((m - M) * LOG2E) + exp2f((x - M) * LOG2E);
        m = M;
    }

    // wave32 butterfly reduction (warpSize == 32 on gfx1250).
    for (int off = 16; off > 0; off >>= 1) {
        float mo = __shfl_xor(m,  off, 32);
        float so = __shfl_xor(s,  off, 32);
        float xo = __shfl_xor(sx, off, 32);
        combine_ms(m, s, mo, so);
        sx += xo;
    }

    // Cross-wave reduction via LDS (8 waves per block).
    __shared__ float sm[8], ss[8], ssx[8];
    const int wave = tid >> 5;
    const int lane = tid & 31;
    if (lane == 0) { sm[wave] = m; ss[wave] = s; ssx[wave] = sx; }
    __syncthreads();

    if (tid == 0) {
        const int nwaves = nthr >> 5;
        for (int w = 1; w < nwaves; ++w) {
            combine_ms(m, s, sm[w], ss[w]);
            sx += ssx[w];
        }
        int   t   = targets[row];
        float xt  = rp[t];
        float lse = m + logf(s);
        float Kf  = (float)K;
        per_row[row] = lse - (1.0f - EPS) * xt - (EPS / Kf) * sx;
    }
}

// Deterministic mean over B per-row losses (single block; fixed summation
// order across replays — no float atomics).
__global__ __launch_bounds__(256) void ce_ls_mean_kernel(
    const float* __restrict__ per_row,
    float*       __restrict__ out,
    int B)
{
    float acc = 0.0f;
    for (int i = threadIdx.x; i < B; i += blockDim.x) acc += per_row[i];

    for (int off = 16; off > 0; off >>= 1) acc += __shfl_xor(acc, off, 32);

    __shared__ float sa[8];
    const int wave = threadIdx.x >> 5;
    const int lane = threadIdx.x & 31;
    if (lane == 0) sa[wave] = acc;
    __syncthreads();

    if (threadIdx.x == 0) {
        float tot = 0.0f;
        for (int w = 0; w < 8; ++w) tot += sa[w];
        out[0] = tot / (float)B;
    }
}

extern "C" void kernel_launch(void* const* d_in, const int* in_sizes, int n_in,
                              void* d_out, int out_size, void* d_ws, size_t ws_size,
                              hipStream_t stream) {
    const float* inputs  = (const float*)d_in[0];
    const int*   targets = (const int*)d_in[1];

    const int B = in_sizes[1];           // 2048
    const int K = in_sizes[0] / B;       // 50257

    float* per_row = (float*)d_ws;       // B floats of scratch

    ce_ls_rows_kernel<<<B, 256, 0, stream>>>(inputs, targets, per_row, K);
    ce_ls_mean_kernel<<<1, 256, 0, stream>>>(per_row, (float*)d_out, B);
}